// SynapticPruner_49065706389536
// MI455X (gfx1250) — compile-verified
//
#include <hip/hip_runtime.h>
#include <stdint.h>

#define THREADS    1024
#define CHUNK      4096          // floats per TDM tile (16 KB)
#define CBINS      8192          // coarse bins = top 13 bits of 31-bit pattern
#define FBINS      262144        // fine bins   = low 18 bits
#define CAP        262144u       // candidate capacity (2^18)
#define SORT_CHUNK 2048

typedef unsigned int       u32;
typedef unsigned long long u64;

typedef unsigned int u32x4 __attribute__((ext_vector_type(4)));
typedef int          i32x8 __attribute__((ext_vector_type(8)));
typedef int          i32x4 __attribute__((ext_vector_type(4)));
typedef float        f32x4 __attribute__((ext_vector_type(4)));

#if defined(__gfx1250__) && __has_builtin(__builtin_amdgcn_tensor_load_to_lds) && __has_builtin(__builtin_amdgcn_s_wait_tensorcnt)
#define USE_TDM 1
#else
#define USE_TDM 0
#endif

__device__ __forceinline__ u32 fbits(float f) { return __float_as_uint(f); }

// meta[] scalars: [0]=b1 coarse weight bucket, [1]=r1 residual rank,
// [2]=threshold bits, [3]=b2 score coarse bucket, [4]=candidate counter

// ---------- Pass 1: coarse histogram of |w| ----------
__global__ void k_hist_w_coarse(const float4* __restrict__ w4, size_t n4,
                                const float* __restrict__ w, size_t n,
                                u32* __restrict__ gA) {
  __shared__ u32 h[CBINS];
  for (u32 b = threadIdx.x; b < CBINS; b += THREADS) h[b] = 0;
  __syncthreads();
  size_t gtid = (size_t)blockIdx.x * blockDim.x + threadIdx.x;
  size_t gs   = (size_t)gridDim.x * blockDim.x;
  for (size_t i = gtid; i < n4; i += gs) {
    if (i + gs < n4) __builtin_prefetch(&w4[i + gs], 0, 0);
    float4 v = w4[i];
    atomicAdd(&h[(fbits(v.x) & 0x7FFFFFFFu) >> 18], 1u);
    atomicAdd(&h[(fbits(v.y) & 0x7FFFFFFFu) >> 18], 1u);
    atomicAdd(&h[(fbits(v.z) & 0x7FFFFFFFu) >> 18], 1u);
    atomicAdd(&h[(fbits(v.w) & 0x7FFFFFFFu) >> 18], 1u);
  }
  for (size_t i = n4 * 4 + gtid; i < n; i += gs)
    atomicAdd(&h[(fbits(w[i]) & 0x7FFFFFFFu) >> 18], 1u);
  __syncthreads();
  for (u32 b = threadIdx.x; b < CBINS; b += THREADS)
    if (h[b]) atomicAdd(&gA[b], h[b]);
}

// ---------- Select coarse bucket containing rank k ----------
__global__ void k_select_w_coarse(const u32* __restrict__ gA, u32 kRank,
                                  u32* __restrict__ meta) {
  __shared__ u32 scan[THREADS];
  const u32 PER = CBINS / THREADS;
  u32 t = threadIdx.x;
  u32 s = 0;
  for (u32 i = 0; i < PER; ++i) s += gA[t * PER + i];
  scan[t] = s; __syncthreads();
  for (u32 off = 1; off < THREADS; off <<= 1) {
    u32 v = (t >= off) ? scan[t - off] : 0;
    __syncthreads();
    scan[t] += v;
    __syncthreads();
  }
  u32 incl = scan[t], before = incl - s;
  if (kRank >= before && kRank < incl) {
    u32 cum = before;
    for (u32 i = 0; i < PER; ++i) {
      u32 hh = gA[t * PER + i];
      if (kRank < cum + hh) { meta[0] = t * PER + i; meta[1] = kRank - cum; break; }
      cum += hh;
    }
  }
}

// ---------- Pass 2: fine histogram inside coarse bucket ----------
__global__ void k_hist_w_fine(const float4* __restrict__ w4, size_t n4,
                              const float* __restrict__ w, size_t n,
                              const u32* __restrict__ meta, u32* __restrict__ gB) {
  u32 b1 = meta[0];
  size_t gtid = (size_t)blockIdx.x * blockDim.x + threadIdx.x;
  size_t gs   = (size_t)gridDim.x * blockDim.x;
  for (size_t i = gtid; i < n4; i += gs) {
    if (i + gs < n4) __builtin_prefetch(&w4[i + gs], 0, 0);
    float4 v = w4[i];
    u32 a;
    a = fbits(v.x) & 0x7FFFFFFFu; if ((a >> 18) == b1) atomicAdd(&gB[a & 0x3FFFFu], 1u);
    a = fbits(v.y) & 0x7FFFFFFFu; if ((a >> 18) == b1) atomicAdd(&gB[a & 0x3FFFFu], 1u);
    a = fbits(v.z) & 0x7FFFFFFFu; if ((a >> 18) == b1) atomicAdd(&gB[a & 0x3FFFFu], 1u);
    a = fbits(v.w) & 0x7FFFFFFFu; if ((a >> 18) == b1) atomicAdd(&gB[a & 0x3FFFFu], 1u);
  }
  for (size_t i = n4 * 4 + gtid; i < n; i += gs) {
    u32 a = fbits(w[i]) & 0x7FFFFFFFu;
    if ((a >> 18) == b1) atomicAdd(&gB[a & 0x3FFFFu], 1u);
  }
}

// ---------- Select fine bucket -> exact threshold bits ----------
__global__ void k_select_w_fine(const u32* __restrict__ gB, u32* __restrict__ meta) {
  __shared__ u32 scan[THREADS];
  const u32 PER = FBINS / THREADS;
  u32 t = threadIdx.x;
  u32 r1 = meta[1];
  u32 s = 0;
  for (u32 i = 0; i < PER; ++i) s += gB[t * PER + i];
  scan[t] = s; __syncthreads();
  for (u32 off = 1; off < THREADS; off <<= 1) {
    u32 v = (t >= off) ? scan[t - off] : 0;
    __syncthreads();
    scan[t] += v;
    __syncthreads();
  }
  u32 incl = scan[t], before = incl - s;
  if (r1 >= before && r1 < incl) {
    u32 cum = before;
    for (u32 i = 0; i < PER; ++i) {
      u32 hh = gB[t * PER + i];
      if (r1 < cum + hh) { meta[2] = (meta[0] << 18) | (t * PER + i); break; }
      cum += hh;
    }
  }
}

#if USE_TDM
// Issue one TDM descriptor: DMA CHUNK floats from global into LDS at lds_off.
__device__ __forceinline__ void tdm_load_tile(const float* gptr, u32 lds_off) {
  unsigned long long ga = (unsigned long long)(const void*)gptr;
  u32x4 g0 = { 1u,                                              // count=1 (valid D#)
               lds_off,                                         // LDS byte address
               (u32)(ga & 0xFFFFFFFFu),                         // global_addr[31:0]
               (u32)((ga >> 32) & 0x01FFFFFFu) | 0x80000000u }; // addr[56:32] | type=2
  i32x8 g1 = { (int)(2u << 16),        // data_size = 4 bytes
               (int)(CHUNK << 16),     // tensor_dim0[15:0] @ bits 63:48
               (int)(1u << 16),        // tensor_dim1 = 1
               (int)(CHUNK << 16),     // tile_dim0 @ bits 127:112
               1,                      // tile_dim1 = 1
               (int)CHUNK,             // tensor_dim0_stride
               0, 0 };
  i32x4 z4 = { 0, 0, 0, 0 };
#if __clang_major__ >= 23
  i32x8 z8 = { 0, 0, 0, 0, 0, 0, 0, 0 };
  __builtin_amdgcn_tensor_load_to_lds(g0, g1, z4, z4, z8, 0);
#else
  __builtin_amdgcn_tensor_load_to_lds(g0, g1, z4, z4, 0);
#endif
}
#endif

// Process one float4: prune + LDS score histogram for masked lanes.
__device__ __forceinline__ void prune4(float4 v, float4 s4, u32 thr,
                                       u32* hist, float4& r) {
  u32 a;
  a = fbits(v.x) & 0x7FFFFFFFu;
  if (a >= thr) r.x = v.x; else { r.x = 0.0f; atomicAdd(&hist[fbits(s4.x) >> 18], 1u); }
  a = fbits(v.y) & 0x7FFFFFFFu;
  if (a >= thr) r.y = v.y; else { r.y = 0.0f; atomicAdd(&hist[fbits(s4.y) >> 18], 1u); }
  a = fbits(v.z) & 0x7FFFFFFFu;
  if (a >= thr) r.z = v.z; else { r.z = 0.0f; atomicAdd(&hist[fbits(s4.z) >> 18], 1u); }
  a = fbits(v.w) & 0x7FFFFFFFu;
  if (a >= thr) r.w = v.w; else { r.w = 0.0f; atomicAdd(&hist[fbits(s4.w) >> 18], 1u); }
}

// ---------- Fused prune + score coarse histogram (double-buffered TDM) ----------
__global__ void k_prune_scorehist(const float* __restrict__ w,
                                  const float4* __restrict__ scores4,
                                  float4* __restrict__ out4,
                                  size_t n,
                                  const u32* __restrict__ meta,
                                  u32* __restrict__ gS) {
  __shared__ __align__(16) u32 smem[2 * CHUNK + CBINS]; // 2x16KB tiles + 32KB hist
  float* tiles = (float*)smem;
  u32*   hist  = smem + 2 * CHUNK;
  for (u32 b = threadIdx.x; b < CBINS; b += THREADS) hist[b] = 0;
  __syncthreads();
  u32 thr = meta[2];
  size_t nChunks = n / CHUNK;
#if USE_TDM
  size_t c0 = blockIdx.x;
  if (c0 < nChunks && threadIdx.x < 32)
    tdm_load_tile(w + c0 * CHUNK, 0u);            // prologue: tile 0 in flight
  u32 buf = 0;
  for (size_t c = c0; c < nChunks; c += gridDim.x) {
    size_t cn = c + gridDim.x;
    if (threadIdx.x < 32) {                        // wave 0 drives the pipeline
      if (cn < nChunks) {
        tdm_load_tile(w + cn * CHUNK, (buf ^ 1u) * (CHUNK * 4u));
        __builtin_amdgcn_s_wait_tensorcnt(1);      // oldest (current tile) done
      } else {
        __builtin_amdgcn_s_wait_tensorcnt(0);      // drain
      }
    }
    __syncthreads();
    size_t i4 = (c * CHUNK >> 2) + threadIdx.x;
    float4 v  = ((const float4*)(tiles + (size_t)buf * CHUNK))[threadIdx.x];
    float4 s4 = scores4[i4];
    float4 r;
    prune4(v, s4, thr, hist, r);
    f32x4 rv = { r.x, r.y, r.z, r.w };
    __builtin_nontemporal_store(rv, (f32x4*)&out4[i4]);  // write-once stream
    __syncthreads();                               // tile free for next DMA
    buf ^= 1u;
  }
#else
  for (size_t c = blockIdx.x; c < nChunks; c += gridDim.x) {
    size_t base = c * (size_t)CHUNK;
    ((float4*)tiles)[threadIdx.x] = ((const float4*)w)[(base >> 2) + threadIdx.x];
    __syncthreads();
    size_t i4 = (base >> 2) + threadIdx.x;
    float4 v  = ((const float4*)tiles)[threadIdx.x];
    float4 s4 = scores4[i4];
    float4 r;
    prune4(v, s4, thr, hist, r);
    f32x4 rv = { r.x, r.y, r.z, r.w };
    __builtin_nontemporal_store(rv, (f32x4*)&out4[i4]);
    __syncthreads();
  }
#endif
  // scalar tail (n not multiple of CHUNK)
  {
    size_t gtid = (size_t)blockIdx.x * blockDim.x + threadIdx.x;
    size_t gs   = (size_t)gridDim.x * blockDim.x;
    float* outf = (float*)out4;
    const float* scf = (const float*)scores4;
    for (size_t i = nChunks * CHUNK + gtid; i < n; i += gs) {
      float v = w[i];
      u32 a = fbits(v) & 0x7FFFFFFFu;
      if (a >= thr) outf[i] = v;
      else { outf[i] = 0.0f; atomicAdd(&hist[fbits(scf[i]) >> 18], 1u); }
    }
  }
  __syncthreads();
  for (u32 b = threadIdx.x; b < CBINS; b += THREADS)
    if (hist[b]) atomicAdd(&gS[b], hist[b]);
}

// ---------- Smallest coarse score bucket with cum >= K ----------
__global__ void k_select_score(const u32* __restrict__ gS, u32 K, u32* __restrict__ meta) {
  __shared__ u32 scan[THREADS];
  const u32 PER = CBINS / THREADS;
  u32 t = threadIdx.x;
  u32 s = 0;
  for (u32 i = 0; i < PER; ++i) s += gS[t * PER + i];
  scan[t] = s; __syncthreads();
  for (u32 off = 1; off < THREADS; off <<= 1) {
    u32 v = (t >= off) ? scan[t - off] : 0;
    __syncthreads();
    scan[t] += v;
    __syncthreads();
  }
  u32 incl = scan[t], before = incl - s;
  if (t == 0) meta[3] = CBINS - 1;   // default: take everything
  __syncthreads();
  if (before < K && incl >= K) {
    u32 cum = before;
    for (u32 i = 0; i < PER; ++i) {
      u32 hh = gS[t * PER + i];
      if (cum < K && cum + hh >= K) { meta[3] = t * PER + i; break; }
      cum += hh;
    }
  }
}

// ---------- Collect candidate (score,index) keys ----------
__global__ void k_collect(const float4* __restrict__ w4, const float4* __restrict__ s4v,
                          size_t n4, const float* __restrict__ w,
                          const float* __restrict__ sf, size_t n,
                          const u32* __restrict__ meta,
                          u64* __restrict__ keys, u32* __restrict__ counter) {
  u32 thr = meta[2], b2 = meta[3];
  size_t gtid = (size_t)blockIdx.x * blockDim.x + threadIdx.x;
  size_t gs   = (size_t)gridDim.x * blockDim.x;
  for (size_t i = gtid; i < n4; i += gs) {
    if (i + gs < n4) { __builtin_prefetch(&w4[i + gs], 0, 0); __builtin_prefetch(&s4v[i + gs], 0, 0); }
    float4 v = w4[i];
    float4 s = s4v[i];
    float vv[4] = { v.x, v.y, v.z, v.w };
    float ss[4] = { s.x, s.y, s.z, s.w };
    #pragma unroll
    for (int c2 = 0; c2 < 4; ++c2) {
      u32 a = fbits(vv[c2]) & 0x7FFFFFFFu;
      if (a < thr) {
        u32 sb = fbits(ss[c2]);
        if ((sb >> 18) <= b2) {
          u32 pos = atomicAdd(counter, 1u);
          if (pos < CAP) keys[pos] = ((u64)sb << 32) | (u32)(i * 4 + c2);
        }
      }
    }
  }
  for (size_t i = n4 * 4 + gtid; i < n; i += gs) {
    u32 a = fbits(w[i]) & 0x7FFFFFFFu;
    if (a < thr) {
      u32 sb = fbits(sf[i]);
      if ((sb >> 18) <= b2) {
        u32 pos = atomicAdd(counter, 1u);
        if (pos < CAP) keys[pos] = ((u64)sb << 32) | (u32)i;
      }
    }
  }
}

// ---------- Pad unused key slots ----------
__global__ void k_pad(u64* __restrict__ keys, const u32* __restrict__ meta) {
  u32 i = blockIdx.x * blockDim.x + threadIdx.x;
  u32 c = meta[4]; if (c > CAP) c = CAP;
  if (i < CAP && i >= c) keys[i] = ~0ull;
}

// ---------- Bitonic sort: local full sort of 2048-key chunks ----------
__global__ void k_sort_local(u64* __restrict__ keys) {
  __shared__ u64 sh[SORT_CHUNK];
  u32 t = threadIdx.x;
  size_t base = (size_t)blockIdx.x * SORT_CHUNK;
  sh[t]           = keys[base + t];
  sh[t + THREADS] = keys[base + t + THREADS];
  __syncthreads();
  for (u32 k2 = 2; k2 <= SORT_CHUNK; k2 <<= 1) {
    for (u32 j = k2 >> 1; j >= 1; j >>= 1) {
      u32 i = (t / j) * (j * 2) + (t % j);
      u32 p = i + j;
      bool up = (((base + i) & (size_t)k2) == 0);
      u64 a = sh[i], b = sh[p];
      if ((a > b) == up) { sh[i] = b; sh[p] = a; }
      __syncthreads();
    }
  }
  keys[base + t]           = sh[t];
  keys[base + t + THREADS] = sh[t + THREADS];
}

// ---------- Bitonic sort: one global compare-exchange step ----------
__global__ void k_sort_global(u64* __restrict__ keys, u32 j, u32 k2) {
  u32 i = blockIdx.x * blockDim.x + threadIdx.x;
  u32 p = i ^ j;
  if (p > i && p < CAP) {
    bool up = ((i & k2) == 0);
    u64 a = keys[i], b = keys[p];
    if ((a > b) == up) { keys[i] = b; keys[p] = a; }
  }
}

// ---------- Bitonic sort: local merge for strides <= 1024 ----------
__global__ void k_merge_local(u64* __restrict__ keys, u32 k2) {
  __shared__ u64 sh[SORT_CHUNK];
  u32 t = threadIdx.x;
  size_t base = (size_t)blockIdx.x * SORT_CHUNK;
  sh[t]           = keys[base + t];
  sh[t + THREADS] = keys[base + t + THREADS];
  __syncthreads();
  for (u32 j = SORT_CHUNK / 2; j >= 1; j >>= 1) {
    u32 i = (t / j) * (j * 2) + (t % j);
    u32 p = i + j;
    bool up = (((base + i) & (size_t)k2) == 0);
    u64 a = sh[i], b = sh[p];
    if ((a > b) == up) { sh[i] = b; sh[p] = a; }
    __syncthreads();
  }
  keys[base + t]           = sh[t];
  keys[base + t + THREADS] = sh[t + THREADS];
}

// ---------- Rank-ordered scatter of regrow values ----------
__global__ void k_scatter(const u64* __restrict__ keys, const float* __restrict__ vals,
                          float* __restrict__ out, u32 K) {
  u32 j = blockIdx.x * blockDim.x + threadIdx.x;
  if (j < K) {
    u64 key = keys[j];
    if (key != ~0ull) out[(u32)key] = vals[j] * 0.01f;
  }
}

extern "C" void kernel_launch(void* const* d_in, const int* in_sizes, int n_in,
                              void* d_out, int out_size, void* d_ws, size_t ws_size,
                              hipStream_t stream) {
  (void)n_in; (void)out_size; (void)ws_size;
  const float* w      = (const float*)d_in[0];
  const float* scores = (const float*)d_in[1];
  const float* vals   = (const float*)d_in[2];
  size_t n  = (size_t)in_sizes[0];
  u32    K  = (u32)in_sizes[2];
  u32 kRank = (u32)((double)n * 0.9);   // matches Python int(n * 0.9)

  char* ws = (char*)d_ws;
  u32* gA   = (u32*)ws;                                        // 32 KB
  u32* gB   = (u32*)(ws + 32768);                              // 1 MB
  u32* gS   = (u32*)(ws + 32768 + 1048576);                    // 32 KB
  u32* meta = (u32*)(ws + 32768 + 1048576 + 32768);            // 64 B
  u64* keys = (u64*)(ws + 32768 + 1048576 + 32768 + 64);       // 2 MB

  hipMemsetAsync(gA,   0, CBINS * sizeof(u32), stream);
  hipMemsetAsync(gB,   0, FBINS * sizeof(u32), stream);
  hipMemsetAsync(gS,   0, CBINS * sizeof(u32), stream);
  hipMemsetAsync(meta, 0, 64, stream);

  size_t n4 = n >> 2;
  dim3 big(2048), blk(THREADS);

  k_hist_w_coarse<<<big, blk, 0, stream>>>((const float4*)w, n4, w, n, gA);
  k_select_w_coarse<<<1, blk, 0, stream>>>(gA, kRank, meta);
  k_hist_w_fine<<<big, blk, 0, stream>>>((const float4*)w, n4, w, n, meta, gB);
  k_select_w_fine<<<1, blk, 0, stream>>>(gB, meta);
  k_prune_scorehist<<<big, blk, 0, stream>>>(w, (const float4*)scores, (float4*)d_out,
                                             n, meta, gS);
  k_select_score<<<1, blk, 0, stream>>>(gS, K, meta);
  k_collect<<<big, blk, 0, stream>>>((const float4*)w, (const float4*)scores, n4,
                                     w, scores, n, meta, keys, meta + 4);
  k_pad<<<CAP / 256, 256, 0, stream>>>(keys, meta);

  k_sort_local<<<CAP / SORT_CHUNK, THREADS, 0, stream>>>(keys);
  for (u32 k2 = 2 * SORT_CHUNK; k2 <= CAP; k2 <<= 1) {
    for (u32 j = k2 >> 1; j >= SORT_CHUNK; j >>= 1)
      k_sort_global<<<CAP / 256, 256, 0, stream>>>(keys, j, k2);
    k_merge_local<<<CAP / SORT_CHUNK, THREADS, 0, stream>>>(keys, k2);
  }

  k_scatter<<<(K + 255) / 256, 256, 0, stream>>>(keys, vals, (float*)d_out, K);
}